// PlanetoidGIN_50465865728499
// MI455X (gfx1250) — compile-verified
//
#include <hip/hip_runtime.h>

typedef __attribute__((ext_vector_type(2))) float v2f;
typedef __attribute__((ext_vector_type(4))) float v4f;
typedef __attribute__((ext_vector_type(8))) float v8f;

#define N_NODES   50000
#define N_EDGES   800000
#define F_IN      512
#define HID       128
#define N_CLASSES 10
#define N_TARGET  1000

// ---------------------------------------------------------------------------
// GEMM: out[M x ldOut] (cols < Nact written) = A[M x K] @ W[K x Nact]
// f32 WMMA 16x16x4, double-buffered LDS, register-staged global loads.
// Block = 256 threads = 8 wave32s; block tile = 128 x (16*NTILES).
// K must be a multiple of 32 (512 and 128 both are).
// Fragment layouts per CDNA5 ISA 7.12.2:
//   A 16x4 : lanes 0-15 -> {K+0,K+1}, lanes 16-31 -> {K+2,K+3}, M = lane&15
//   B 4x16 : lanes 0-15 -> rows {K+0,K+1}, lanes 16-31 -> rows {K+2,K+3}, N = lane&15
//   C/D    : VGPR j -> M = j (lanes 0-15) / j+8 (lanes 16-31), N = lane&15
// sB stores B row-pairs interleaved: {B[2kp][n], B[2kp+1][n]} contiguous, so a
// lane's whole B fragment is one aligned 8-byte LDS load (ds_load_b64).
// ---------------------------------------------------------------------------
template <int NTILES>
__global__ __launch_bounds__(256)
void gemm_wmma(const float* __restrict__ A, const float* __restrict__ W,
               float* __restrict__ out, int M, int K, int Nact, int ldOut)
{
    constexpr int NPAD     = NTILES * 16;
    constexpr int BPT      = (32 * NPAD) / 256;              // B floats per thread
    constexpr int PADB     = ((16 - (NPAD % 32)) + 32) % 32; // pair-row pad (float2s):
                                                             // makes pair stride == 32 banks mod 64
    constexpr int BSTRIDE2 = NPAD + PADB;                    // float2 stride per k-pair row

    __shared__ float sA[2][128 * 34];                        // 128 rows x 32 k, stride 34
    __shared__ float sB[2][16 * BSTRIDE2 * 2];               // 16 k-pairs x NPAD float2s

    const int tid     = threadIdx.x;
    const int wave    = tid >> 5;
    const int lane    = tid & 31;
    const int half    = lane >> 4;          // 0: lanes 0-15, 1: lanes 16-31
    const int m       = lane & 15;
    const int rowBase = blockIdx.x * 128;

    // ---- per-thread staging geometry (constant across K chunks) ----
    int          rA[4], kA[4];
    const float* aSrc[4];
#pragma unroll
    for (int v = 0; v < 4; ++v) {
        int fi = tid + v * 256;             // float4 id 0..1023
        rA[v]  = fi >> 3;                   // row within 128-row tile
        kA[v]  = (fi & 7) << 2;             // k offset 0,4,...,28
        int gr = rowBase + rA[v]; if (gr >= M) gr = M - 1;   // clamp (stores guarded)
        aSrc[v] = A + (size_t)gr * K + kA[v];
    }
    int kB[BPT], nB[BPT], ofsB[BPT];
#pragma unroll
    for (int j = 0; j < BPT; ++j) {
        int i   = tid + j * 256;
        kB[j]   = i / NPAD;
        nB[j]   = i % NPAD;
        ofsB[j] = (kB[j] >> 1) * (2 * BSTRIDE2) + 2 * nB[j] + (kB[j] & 1);
    }

    v8f acc[NTILES];
    v8f zero = {};
#pragma unroll
    for (int t = 0; t < NTILES; ++t) acc[t] = zero;

    v4f   aR[4];
    float bR[BPT];

    auto stageRegs = [&](int k0) {
#pragma unroll
        for (int v = 0; v < 4; ++v)
            aR[v] = __builtin_nontemporal_load((const v4f*)(aSrc[v] + k0));
#pragma unroll
        for (int j = 0; j < BPT; ++j)
            bR[j] = (nB[j] < Nact) ? W[(size_t)(k0 + kB[j]) * Nact + nB[j]] : 0.0f;
    };
    auto storeLDS = [&](int bi) {
#pragma unroll
        for (int v = 0; v < 4; ++v) {
            float* d = &sA[bi][rA[v] * 34 + kA[v]];
            d[0] = aR[v].x; d[1] = aR[v].y; d[2] = aR[v].z; d[3] = aR[v].w;
        }
#pragma unroll
        for (int j = 0; j < BPT; ++j)
            sB[bi][ofsB[j]] = bR[j];
    };

    const int nChunks = K >> 5;

    stageRegs(0);
    storeLDS(0);
    __syncthreads();

    for (int c = 0; c < nChunks; ++c) {
        const int  bi      = c & 1;
        const bool hasNext = (c + 1) < nChunks;
        if (hasNext) stageRegs((c + 1) << 5);        // global loads overlap WMMA below

        const float* aPtr = &sA[bi][(wave * 16 + m) * 34 + 2 * half];
        const float* bPtr = &sB[bi][half * (2 * BSTRIDE2) + 2 * m];
#pragma unroll
        for (int kk = 0; kk < 32; kk += 4) {
            v2f a = *(const v2f*)(aPtr + kk);
            const float* bb = bPtr + (kk >> 1) * (2 * BSTRIDE2);
            v2f bfrag[NTILES];
#pragma unroll
            for (int t = 0; t < NTILES; ++t)
                bfrag[t] = *(const v2f*)(bb + 32 * t);     // one ds_load_b64 each
#pragma unroll
            for (int t = 0; t < NTILES; ++t)
                acc[t] = __builtin_amdgcn_wmma_f32_16x16x4_f32(
                    /*neg_a=*/false, a, /*neg_b=*/false, bfrag[t],
                    /*c_mod=*/(short)0, acc[t],
                    /*reuse_a=*/false, /*reuse_b=*/false);
        }

        if (hasNext) storeLDS((c + 1) & 1);
        __syncthreads();
    }

    // ---- store C/D tiles ----
#pragma unroll
    for (int t = 0; t < NTILES; ++t) {
#pragma unroll
        for (int j = 0; j < 8; ++j) {
            int row = rowBase + wave * 16 + half * 8 + j;
            int col = t * 16 + m;
            if (row < M && col < Nact)
                out[(size_t)row * ldOut + col] = acc[t][j];
        }
    }
}

// ---------------------------------------------------------------------------
// Edge aggregation in projected 128-dim space: agg[dst] += w * p[src]
// One wave per edge; lane covers 4 consecutive floats (float4 gather).
// ---------------------------------------------------------------------------
__global__ __launch_bounds__(256)
void scatter128(const float* __restrict__ p, const int* __restrict__ ei,
                const float* __restrict__ ew, float* __restrict__ agg)
{
    int wave = threadIdx.x >> 5, lane = threadIdx.x & 31;
    int e = blockIdx.x * 8 + wave;
    if (e >= N_EDGES) return;
    int src = ei[e];
    int dst = ei[N_EDGES + e];
    float w = ew[e];
    v4f v = *(const v4f*)(p + (size_t)src * HID + lane * 4);
    float* o = agg + (size_t)dst * HID + lane * 4;
    unsafeAtomicAdd(o + 0, w * v.x);
    unsafeAtomicAdd(o + 1, w * v.y);
    unsafeAtomicAdd(o + 2, w * v.z);
    unsafeAtomicAdd(o + 3, w * v.w);
}

// Layer-3 aggregation in 10-dim (stride-16) space.
__global__ __launch_bounds__(256)
void scatter10(const float* __restrict__ p, const int* __restrict__ ei,
               const float* __restrict__ ew, float* __restrict__ agg)
{
    int idx = blockIdx.x * 256 + threadIdx.x;
    int e = idx >> 4, f = idx & 15;
    if (e >= N_EDGES || f >= N_CLASSES) return;
    int src = ei[e];
    int dst = ei[N_EDGES + e];
    unsafeAtomicAdd(agg + (size_t)dst * 16 + f, ew[e] * p[(size_t)src * 16 + f]);
}

// In-place epilogue: p = relu(p + agg + b), feature dim 128, float4-vectorized.
__global__ __launch_bounds__(256)
void add_bias_relu128(float* __restrict__ p, const float* __restrict__ agg,
                      const float* __restrict__ b)
{
    int i = blockIdx.x * 256 + threadIdx.x;          // float4 index
    if (i >= N_NODES * (HID / 4)) return;
    int c4 = i & (HID / 4 - 1);
    v4f pv = ((const v4f*)p)[i];
    v4f av = ((const v4f*)agg)[i];
    v4f bv = ((const v4f*)b)[c4];
    v4f r;
    r.x = fmaxf(pv.x + av.x + bv.x, 0.0f);
    r.y = fmaxf(pv.y + av.y + bv.y, 0.0f);
    r.z = fmaxf(pv.z + av.z + bv.z, 0.0f);
    r.w = fmaxf(pv.w + av.w + bv.w, 0.0f);
    ((v4f*)p)[i] = r;
}

// Final: out[t][c] = p3[target[t]][c] + agg3[target[t]][c] + b3[c]  (no relu)
__global__ __launch_bounds__(256)
void final_gather(const float* __restrict__ p3, const float* __restrict__ agg3,
                  const float* __restrict__ b3, const int* __restrict__ tgt,
                  float* __restrict__ out)
{
    int i = blockIdx.x * 256 + threadIdx.x;
    if (i >= N_TARGET * N_CLASSES) return;
    int t = i / N_CLASSES, c = i % N_CLASSES;
    int node = tgt[t];
    out[i] = p3[(size_t)node * 16 + c] + agg3[(size_t)node * 16 + c] + b3[c];
}

// ---------------------------------------------------------------------------
extern "C" void kernel_launch(void* const* d_in, const int* in_sizes, int n_in,
                              void* d_out, int out_size, void* d_ws, size_t ws_size,
                              hipStream_t stream)
{
    const float* x   = (const float*)d_in[0];
    const int*   ei  = (const int*)  d_in[1];   // [2, E] flat
    const float* ew  = (const float*)d_in[2];
    const int*   tgt = (const int*)  d_in[3];
    const float* W1  = (const float*)d_in[4];
    const float* b1  = (const float*)d_in[5];
    const float* W2  = (const float*)d_in[6];
    const float* b2  = (const float*)d_in[7];
    const float* W3  = (const float*)d_in[8];
    const float* b3  = (const float*)d_in[9];
    float* outp = (float*)d_out;

    const size_t NH  = (size_t)N_NODES * HID;   // 6.4M floats
    const size_t N16 = (size_t)N_NODES * 16;    // 0.8M floats
    float* buf0 = (float*)d_ws;      // p1 / h1   [N,128]
    float* buf1 = buf0 + NH;         // agg       [N,128]
    float* buf2 = buf1 + NH;         // p2 / h2   [N,128]
    float* buf3 = buf2 + NH;         // p3        [N,16] (cols 0-9 valid)
    float* buf4 = buf3 + N16;        // agg3      [N,16]

    const int gemmBlocks = (N_NODES + 127) / 128;         // 391
    const int edgeBlocks = (N_EDGES + 7) / 8;             // wave-per-edge
    const int eluBlocks  = (N_NODES * (HID / 4) + 255) / 256;

    // ---- Layer 1: project first (linearity of segment_sum), then aggregate ----
    gemm_wmma<8><<<gemmBlocks, 256, 0, stream>>>(x, W1, buf0, N_NODES, F_IN, HID, HID);
    (void)hipMemsetAsync(buf1, 0, NH * sizeof(float), stream);
    scatter128<<<edgeBlocks, 256, 0, stream>>>(buf0, ei, ew, buf1);
    add_bias_relu128<<<eluBlocks, 256, 0, stream>>>(buf0, buf1, b1);

    // ---- Layer 2 ----
    gemm_wmma<8><<<gemmBlocks, 256, 0, stream>>>(buf0, W2, buf2, N_NODES, HID, HID, HID);
    (void)hipMemsetAsync(buf1, 0, NH * sizeof(float), stream);
    scatter128<<<edgeBlocks, 256, 0, stream>>>(buf2, ei, ew, buf1);
    add_bias_relu128<<<eluBlocks, 256, 0, stream>>>(buf2, buf1, b2);

    // ---- Layer 3 (N=10 padded to 16) ----
    gemm_wmma<1><<<gemmBlocks, 256, 0, stream>>>(buf2, W3, buf3, N_NODES, HID, N_CLASSES, 16);
    (void)hipMemsetAsync(buf4, 0, N16 * sizeof(float), stream);
    scatter10<<<(N_EDGES * 16 + 255) / 256, 256, 0, stream>>>(buf3, ei, ew, buf4);
    final_gather<<<(N_TARGET * N_CLASSES + 255) / 256, 256, 0, stream>>>(buf3, buf4, b3, tgt, outp);
}